// SphereUpGeo_72756745994743
// MI455X (gfx1250) — compile-verified
//
#include <hip/hip_runtime.h>
#include <hip/hip_bf16.h>
#include <stdint.h>

// SphereUpGeo: x[1,16,K] -> out[1,16,N], N = 4K.
// Deterministic atomic-free gather exploiting the fixed edge structure:
//   child edge    e = 8p + j (j<4)  -> col = 4p + j
//   neighbor edge e = 8p + 4 + j    -> col = (4p + off) % N, off in [4,64)
// Per fine pixel f the only candidate sources are p0 = f>>2 (child) and the
// 15 coarse pixels q = (f-d)>>2 with d = 4+(f&3)+4k. Each 256-thread block
// touches a fixed 80-coarse-pixel window; edges + x window are staged into
// LDS via CDNA5 async global->LDS loads (ASYNCcnt).

#define KOUT   786432
#define NIN    3145728
#define CH     16
#define BLK    256
#define NQ     80            // constant coarse-pixel window per block
#define NEDGE  (NQ * 8)      // 640 edges
#define NX     (NQ * CH)     // 1280 staged x values

// Builtin prototype (hipcc diag round 1): typed int* in AS1 ("__device__") / AS3 ("__shared__").
typedef __attribute__((address_space(1))) int g_i32;
typedef __attribute__((address_space(3))) int l_i32;

#if defined(__AMDGCN__) && __has_builtin(__builtin_amdgcn_global_load_async_to_lds_b32)
#define USE_ASYNC 1
#else
#define USE_ASYNC 0
#endif

__global__ __launch_bounds__(BLK) void sphere_up_gather(
    const float* __restrict__ x,      // [CH, KOUT]
    const float* __restrict__ evals,  // [KOUT*8]
    const int*   __restrict__ ecols,  // [KOUT*8]
    float* __restrict__ out)          // [CH, NIN]
{
  __shared__ int   s_cols[NEDGE];
  __shared__ float s_vals[NEDGE];
  __shared__ float s_x[NX];           // layout [c][q_local]

  const int tid = threadIdx.x;
  const int F0  = blockIdx.x * BLK;
  const int f   = F0 + tid;

  // Constant-width window: qmin = max(0, F0/4 - 16); covers all candidates.
  int qmin = (F0 >> 2) - 16; if (qmin < 0) qmin = 0;
  const int eb = qmin * 8;

#if USE_ASYNC
#pragma unroll
  for (int i = tid; i < NEDGE; i += BLK) {
    __builtin_amdgcn_global_load_async_to_lds_b32(
        (g_i32*)(ecols + eb + i), (l_i32*)(&s_cols[i]), 0, 0);
    __builtin_amdgcn_global_load_async_to_lds_b32(
        (g_i32*)(evals + eb + i), (l_i32*)(&s_vals[i]), 0, 0);
  }
#pragma unroll
  for (int i = tid; i < NX; i += BLK) {        // 5 full-EXEC iterations
    const int c  = i / NQ;                     // constant divisor -> cheap
    const int ql = i - c * NQ;
    __builtin_amdgcn_global_load_async_to_lds_b32(
        (g_i32*)(x + c * KOUT + qmin + ql), (l_i32*)(&s_x[i]), 0, 0);
  }
#if __has_builtin(__builtin_amdgcn_s_wait_asynccnt)
  __builtin_amdgcn_s_wait_asynccnt(0);
#else
  asm volatile("s_wait_asynccnt 0" ::: "memory");
#endif
#else
#pragma unroll
  for (int i = tid; i < NEDGE; i += BLK) {
    s_cols[i] = ecols[eb + i];
    s_vals[i] = evals[eb + i];
  }
#pragma unroll
  for (int i = tid; i < NX; i += BLK) {
    const int c  = i / NQ;
    const int ql = i - c * NQ;
    s_x[i] = x[c * KOUT + qmin + ql];
  }
#endif
  __syncthreads();

  float acc[CH];
#pragma unroll
  for (int c = 0; c < CH; ++c) acc[c] = 0.0f;
  float den = 0.0f;
  const int r = f & 3;

  // Child edge: e = 8*(f>>2) + (f&3), col == f by construction.
  {
    const int p0 = f >> 2;
    const int ql = p0 - qmin;
    const float w = s_vals[p0 * 8 - eb + r];
    den += w;
#pragma unroll
    for (int c = 0; c < CH; ++c) acc[c] += w * s_x[c * NQ + ql];
  }

  // Neighbor edges: 15 candidate coarse pixels, 4 edges each.
  for (int k = 0; k < 15; ++k) {
    const int d = 4 + r + 4 * k;
    int s = f - d;
    if (s >= 0) {
      const int q  = s >> 2;             // q in [qmin, qmin+79] by construction
      const int ql = q - qmin;
      const int li = q * 8 - eb + 4;
#pragma unroll
      for (int j = 0; j < 4; ++j) {
        if (s_cols[li + j] == f) {
          const float w = s_vals[li + j];
          den += w;
#pragma unroll
          for (int c = 0; c < CH; ++c) acc[c] += w * s_x[c * NQ + ql];
        }
      }
    } else {
      // Wrap-around: only threads with f < 63 in block 0 ever take this.
      s += NIN;
      const int q = s >> 2;
      const int e = q * 8 + 4;
#pragma unroll
      for (int j = 0; j < 4; ++j) {
        if (ecols[e + j] == f) {
          const float w = evals[e + j];
          den += w;
#pragma unroll
          for (int c = 0; c < CH; ++c) acc[c] += w * x[c * KOUT + q];
        }
      }
    }
  }

  // den == col_sum[f] exactly (all edges targeting f were enumerated).
  const float rd = 1.0f / fmaxf(den, 1e-12f);
#pragma unroll
  for (int c = 0; c < CH; ++c) out[c * NIN + f] = acc[c] * rd;
}

extern "C" void kernel_launch(void* const* d_in, const int* in_sizes, int n_in,
                              void* d_out, int out_size, void* d_ws, size_t ws_size,
                              hipStream_t stream) {
  (void)in_sizes; (void)n_in; (void)out_size; (void)d_ws; (void)ws_size;
  const float* x     = (const float*)d_in[0];   // [1,16,KOUT] f32
  const float* evals = (const float*)d_in[1];   // [KOUT*8]    f32
  /* d_in[2] = edge_rows : implied by edge index, unused */
  const int*   ecols = (const int*)d_in[3];     // [KOUT*8]    int
  float* out = (float*)d_out;                   // [1,16,NIN]  f32

  dim3 grid(NIN / BLK);
  dim3 block(BLK);
  hipLaunchKernelGGL(sphere_up_gather, grid, block, 0, stream, x, evals, ecols, out);
}